// FeatureExtraction_91302414778570
// MI455X (gfx1250) — compile-verified
//
#include <hip/hip_runtime.h>
#include <hip/hip_bf16.h>
#include <math.h>

#define DIM 64
#define NUM_LAYERS 3
#define NEG_SLOPE 0.2f

typedef __attribute__((ext_vector_type(2))) float v2f;
typedef __attribute__((ext_vector_type(8))) float v8f;

// ---- order-preserving float <-> uint key for atomicMax-based segment_max ----
__device__ __forceinline__ unsigned fkey(float f) {
    unsigned u = __float_as_uint(f);
    return (u & 0x80000000u) ? ~u : (u | 0x80000000u);
}
__device__ __forceinline__ float fkey_inv(unsigned k) {
    unsigned u = (k & 0x80000000u) ? (k & 0x7FFFFFFFu) : ~k;
    return __uint_as_float(u);
}
__device__ __forceinline__ float lrelu(float v) {
    return v > 0.0f ? v : NEG_SLOPE * v;
}

// ---------------------------------------------------------------------------
// h = x @ W  via V_WMMA_F32_16X16X4_F32. One wave -> 16 rows x 64 cols.
// A: lane l holds A[M = l&15, K = 2*(l>>4)+v]; B mirrored; D per ISA layout.
// ---------------------------------------------------------------------------
__global__ void gat_gemm_wmma(const float* __restrict__ x,
                              const float* __restrict__ W,
                              float* __restrict__ h, int n) {
    const int lane = threadIdx.x & 31;
    const int wave = threadIdx.x >> 5;
    const int tile = blockIdx.x * 8 + wave;      // uniform within wave
    const int row_base = tile * 16;
    if (row_base >= n) return;                   // wave-uniform exit: EXEC stays full

    const int m  = lane & 15;
    const int kh = lane >> 4;
    int r = row_base + m;
    if (r >= n) r = n - 1;                       // clamp ragged loads, predicate stores
    const float* xr = x + (long)r * DIM;

    v8f acc0 = {}, acc1 = {}, acc2 = {}, acc3 = {};
    #pragma unroll
    for (int kk = 0; kk < DIM; kk += 4) {
        const int k0 = kk + 2 * kh;
        v2f a; a.x = xr[k0]; a.y = xr[k0 + 1];
        const float* w0 = W + k0 * DIM;          // row k0 of W (row-major [din][dout])
        v2f b0; b0.x = w0[m];      b0.y = w0[DIM + m];
        v2f b1; b1.x = w0[16 + m]; b1.y = w0[DIM + 16 + m];
        v2f b2; b2.x = w0[32 + m]; b2.y = w0[DIM + 32 + m];
        v2f b3; b3.x = w0[48 + m]; b3.y = w0[DIM + 48 + m];
        acc0 = __builtin_amdgcn_wmma_f32_16x16x4_f32(false, a, false, b0, (short)0, acc0, false, false);
        acc1 = __builtin_amdgcn_wmma_f32_16x16x4_f32(false, a, false, b1, (short)0, acc1, false, false);
        acc2 = __builtin_amdgcn_wmma_f32_16x16x4_f32(false, a, false, b2, (short)0, acc2, false, false);
        acc3 = __builtin_amdgcn_wmma_f32_16x16x4_f32(false, a, false, b3, (short)0, acc3, false, false);
    }
    #pragma unroll
    for (int vr = 0; vr < 8; ++vr) {
        int row = row_base + vr + 8 * kh;        // D layout: M = vgpr + 8*(lane>=16)
        if (row < n) {
            float* hr = h + (long)row * DIM + m;
            hr[0]  = acc0[vr];
            hr[16] = acc1[vr];
            hr[32] = acc2[vr];
            hr[48] = acc3[vr];
        }
    }
}

// ---------------------------------------------------------------------------
// Per-node attention scalars: al_src = h.a_src, al_dst = h.a_dst (wave/node)
// ---------------------------------------------------------------------------
__global__ void gat_al(const float* __restrict__ h,
                       const float* __restrict__ asrc,
                       const float* __restrict__ adst,
                       float* __restrict__ als, float* __restrict__ ald, int n) {
    int node = (int)((blockIdx.x * (long)blockDim.x + threadIdx.x) >> 5);
    int lane = threadIdx.x & 31;
    if (node >= n) return;
    const float* hr = h + (long)node * DIM;
    float h0 = hr[lane], h1 = hr[lane + 32];
    float s = h0 * asrc[lane] + h1 * asrc[lane + 32];
    float d = h0 * adst[lane] + h1 * adst[lane + 32];
    #pragma unroll
    for (int off = 16; off; off >>= 1) {
        s += __shfl_xor(s, off, 32);
        d += __shfl_xor(d, off, 32);
    }
    if (lane == 0) { als[node] = s; ald[node] = d; }
}

// ---------------------------------------------------------------------------
// Zero accumulators / max keys / denominators for the layer
// ---------------------------------------------------------------------------
__global__ void gat_init(unsigned* __restrict__ mkey, float* __restrict__ denom,
                         float* __restrict__ acc, int n) {
    long i = blockIdx.x * (long)blockDim.x + threadIdx.x;
    if (i < (long)n * DIM) acc[i] = 0.0f;
    if (i < n) { mkey[i] = 0u; denom[i] = 0.0f; }
}

// ---------------------------------------------------------------------------
// Edge pass 1: logits + segment_max(dst) via uint-key atomicMax
// ---------------------------------------------------------------------------
__global__ void gat_edge_logit(const int* __restrict__ esrc, const int* __restrict__ edst,
                               int E0, int Etot,
                               const float* __restrict__ als, const float* __restrict__ ald,
                               float* __restrict__ logits, unsigned* __restrict__ mkey) {
    int e = blockIdx.x * blockDim.x + threadIdx.x;
    if (e >= Etot) return;
    if (e + 4096 < E0) {                          // global_prefetch_b8, edge list ahead
        __builtin_prefetch(esrc + e + 4096, 0, 1);
        __builtin_prefetch(edst + e + 4096, 0, 1);
    }
    int s, d;
    if (e < E0) { s = esrc[e]; d = edst[e]; } else { s = d = e - E0; }  // self-loops
    float z = lrelu(als[s] + ald[d]);
    logits[e] = z;
    atomicMax(mkey + d, fkey(z));
}

// ---------------------------------------------------------------------------
// Edge pass 2: e = exp(logit - max[dst]); denom[dst] += e  (overwrite logits)
// ---------------------------------------------------------------------------
__global__ void gat_edge_exp(const int* __restrict__ edst, int E0, int Etot,
                             const unsigned* __restrict__ mkey,
                             float* __restrict__ logits, float* __restrict__ denom) {
    int e = blockIdx.x * blockDim.x + threadIdx.x;
    if (e >= Etot) return;
    int d = (e < E0) ? edst[e] : (e - E0);
    float mx = fkey_inv(mkey[d]);
    float w = __expf(logits[e] - mx);
    logits[e] = w;
    atomicAdd(denom + d, w);
}

// ---------------------------------------------------------------------------
// Edge pass 3: acc[dst] += alpha * h[src]. One wave per edge, float2 per lane.
// ---------------------------------------------------------------------------
__global__ void gat_edge_scatter(const int* __restrict__ esrc, const int* __restrict__ edst,
                                 int E0, int Etot,
                                 const float* __restrict__ h, const float* __restrict__ ew,
                                 const float* __restrict__ denom, float* __restrict__ acc) {
    long gid = blockIdx.x * (long)blockDim.x + threadIdx.x;
    int e = (int)(gid >> 5);
    int lane = threadIdx.x & 31;
    if (e >= Etot) return;
    int s, d;
    if (e < E0) { s = esrc[e]; d = edst[e]; } else { s = d = e - E0; }
    float alpha = ew[e] / denom[d];
    const float2* hr = (const float2*)(h + (long)s * DIM);
    float2 v = hr[lane];
    float* ar = acc + (long)d * DIM + lane * 2;
    atomicAdd(ar,     v.x * alpha);
    atomicAdd(ar + 1, v.y * alpha);
}

// ---------------------------------------------------------------------------
// Finalize: out = acc + bias, leaky-relu between layers
// ---------------------------------------------------------------------------
__global__ void gat_finalize(const float* __restrict__ acc, const float* __restrict__ bias,
                             float* __restrict__ out, int n, int do_relu) {
    long i = blockIdx.x * (long)blockDim.x + threadIdx.x;
    if (i >= (long)n * DIM) return;
    float v = acc[i] + bias[i & (DIM - 1)];
    if (do_relu) v = lrelu(v);
    out[i] = v;
}

extern "C" void kernel_launch(void* const* d_in, const int* in_sizes, int n_in,
                              void* d_out, int out_size, void* d_ws, size_t ws_size,
                              hipStream_t stream) {
    (void)n_in; (void)out_size; (void)ws_size;
    const float* x0   = (const float*)d_in[0];    // [4,25000,64] -> [N,64]
    const int*   eidx = (const int*)d_in[1];      // [2,E0]
    const float* Wl   = (const float*)d_in[2];    // [L,64,64]
    const float* asrc = (const float*)d_in[3];    // [L,64]
    const float* adst = (const float*)d_in[4];    // [L,64]
    const float* bias = (const float*)d_in[5];    // [L,64]

    const int n    = in_sizes[0] / DIM;           // 100000
    const int E0   = in_sizes[1] / 2;             // 1200000
    const int Etot = E0 + n;                      // + self loops
    const int* esrc = eidx;
    const int* edst = eidx + E0;

    // workspace carve-up (all fits in L2: ~84 MB total)
    float* ws  = (float*)d_ws;
    float* h    = ws;  ws += (size_t)n * DIM;
    float* accb = ws;  ws += (size_t)n * DIM;
    float* xbuf = ws;  ws += (size_t)n * DIM;
    float* als  = ws;  ws += n;
    float* ald  = ws;  ws += n;
    unsigned* mkey = (unsigned*)ws; ws += n;
    float* denom = ws; ws += n;
    float* ew    = ws; ws += Etot;

    const int TB = 256;
    const int g_gemm = ((n + 15) / 16 + 7) / 8;          // 8 waves (tiles) per block
    const int g_nd   = (n * DIM + TB - 1) / TB;
    const int g_n32  = (n * 32 + TB - 1) / TB;           // wave per node
    const int g_e    = (Etot + TB - 1) / TB;
    const int g_e32  = (int)(((long)Etot * 32 + TB - 1) / TB);  // wave per edge

    for (int L = 0; L < NUM_LAYERS; ++L) {
        const float* xin = (L == 0) ? x0 : xbuf;
        gat_gemm_wmma<<<g_gemm, TB, 0, stream>>>(xin, Wl + L * DIM * DIM, h, n);
        gat_al<<<g_n32, TB, 0, stream>>>(h, asrc + L * DIM, adst + L * DIM, als, ald, n);
        gat_init<<<g_nd, TB, 0, stream>>>(mkey, denom, accb, n);
        gat_edge_logit<<<g_e, TB, 0, stream>>>(esrc, edst, E0, Etot, als, ald, ew, mkey);
        gat_edge_exp<<<g_e, TB, 0, stream>>>(edst, E0, Etot, mkey, ew, denom);
        gat_edge_scatter<<<g_e32, TB, 0, stream>>>(esrc, edst, E0, Etot, h, ew, denom, accb);
        float* outp = (L == NUM_LAYERS - 1) ? (float*)d_out : xbuf;
        gat_finalize<<<g_nd, TB, 0, stream>>>(accb, bias + L * DIM, outp, n, L < NUM_LAYERS - 1);
    }
}